// GATClassifier_19353122636052
// MI455X (gfx1250) — compile-verified
//
#include <hip/hip_runtime.h>
#include <hip/hip_bf16.h>
#include <math.h>

typedef __attribute__((ext_vector_type(2))) float v2f;
typedef __attribute__((ext_vector_type(8))) float v8f;

#define NN   20000
#define NE   320000
#define ET   (NE + NN)          // edges + self loops
#define DIN  256
#define H    4
#define C1   128
#define C2   64
#define HC1  (H * C1)           // 512
#define HC2  (H * C2)           // 256
#define NG   32

// ---------------------------------------------------------------------------
// helpers: monotone float <-> uint mapping for atomic max
// ---------------------------------------------------------------------------
__device__ __forceinline__ unsigned flipf(float f) {
  unsigned u = __float_as_uint(f);
  return (u >> 31) ? ~u : (u | 0x80000000u);
}
__device__ __forceinline__ float unflipf(unsigned k) {
  unsigned u = (k >> 31) ? (k & 0x7fffffffu) : ~k;
  return __uint_as_float(u);
}
__device__ __forceinline__ float elu1(float v) { return v > 0.f ? v : expf(v) - 1.f; }

// ---------------------------------------------------------------------------
// WMMA fp32 GEMM:  C[M,N] = A[M,K] * B[K,N]   (all row-major, M%16==0, K%4==0)
// one wave -> 16 x 64 tile (4 accumulators), 4 waves/block -> 16 x 256
// uses V_WMMA_F32_16X16X4_F32 (full fp32 matrix pipe, no precision loss)
// ---------------------------------------------------------------------------
__global__ void gemm_wmma_f32(const float* __restrict__ A,
                              const float* __restrict__ B,
                              float* __restrict__ C, int M, int K, int N) {
  const int wave = threadIdx.x >> 5;
  const int lane = threadIdx.x & 31;
  const int mBase = blockIdx.x * 16;
  const int nBase = blockIdx.y * 256 + wave * 64;
  if (nBase >= N || mBase >= M) return;   // wave-uniform guard (EXEC stays all-1s)

  const int half = lane >> 4;   // 0: K pair {0,1}, 1: K pair {2,3}
  const int l16  = lane & 15;
  const size_t aRow = (size_t)(mBase + l16) * K;

  v8f acc0 = {}, acc1 = {}, acc2 = {}, acc3 = {};
  for (int k = 0; k < K; k += 4) {
    const int ka = k + half * 2;
    v2f a;
    a.x = A[aRow + ka];
    a.y = A[aRow + ka + 1];
    const float* __restrict__ B0 = B + (size_t)ka * N;   // row ka
    const float* __restrict__ B1 = B0 + N;               // row ka+1
    const int n0 = nBase + l16;
    v2f b0, b1, b2, b3;
    b0.x = B0[n0];      b0.y = B1[n0];
    b1.x = B0[n0 + 16]; b1.y = B1[n0 + 16];
    b2.x = B0[n0 + 32]; b2.y = B1[n0 + 32];
    b3.x = B0[n0 + 48]; b3.y = B1[n0 + 48];
    acc0 = __builtin_amdgcn_wmma_f32_16x16x4_f32(false, a, false, b0, (short)0, acc0, false, false);
    acc1 = __builtin_amdgcn_wmma_f32_16x16x4_f32(false, a, false, b1, (short)0, acc1, false, false);
    acc2 = __builtin_amdgcn_wmma_f32_16x16x4_f32(false, a, false, b2, (short)0, acc2, false, false);
    acc3 = __builtin_amdgcn_wmma_f32_16x16x4_f32(false, a, false, b3, (short)0, acc3, false, false);
  }
  // C/D layout: VGPR v -> row v (lanes 0-15) / row v+8 (lanes 16-31), col = lane%16
#pragma unroll
  for (int v = 0; v < 8; ++v) {
    const int row = mBase + v + half * 8;
    float* crow = C + (size_t)row * N + nBase + l16;
    crow[0]  = acc0[v];
    crow[16] = acc1[v];
    crow[32] = acc2[v];
    crow[48] = acc3[v];
  }
}

// ---------------------------------------------------------------------------
// scalar sum reduction (edge_attr mean) + finalize
// ---------------------------------------------------------------------------
__global__ void reduce_sum(const float* __restrict__ x, float* __restrict__ dst, int n) {
  __shared__ float sh[256];
  float s = 0.f;
  for (int i = blockIdx.x * blockDim.x + threadIdx.x; i < n; i += gridDim.x * blockDim.x)
    s += x[i];
  sh[threadIdx.x] = s; __syncthreads();
  for (int o = 128; o > 0; o >>= 1) {
    if (threadIdx.x < o) sh[threadIdx.x] += sh[threadIdx.x + o];
    __syncthreads();
  }
  if (threadIdx.x == 0) atomicAdd(dst, sh[0]);
}
__global__ void finalize_mean(float* misc, int n) {
  if (blockIdx.x == 0 && threadIdx.x == 0) misc[0] /= (float)n;
}

// ce[h] = dot(We[0, h*C : h*C+C], ae[h, :])   (al_e = ea * ce[h])
__global__ void edge_coef(const float* __restrict__ We, const float* __restrict__ ae,
                          float* __restrict__ dst, int C) {
  __shared__ float sh[128];
  const int hh = blockIdx.x;
  float s = 0.f;
  for (int c = threadIdx.x; c < C; c += blockDim.x)
    s += We[hh * C + c] * ae[hh * C + c];
  sh[threadIdx.x] = s; __syncthreads();
  for (int o = blockDim.x >> 1; o > 0; o >>= 1) {
    if (threadIdx.x < o) sh[threadIdx.x] += sh[threadIdx.x + o];
    __syncthreads();
  }
  if (threadIdx.x == 0) dst[hh] = sh[0];
}

// ---------------------------------------------------------------------------
// al_s / al_d : per (node, head) dot over channels; one wave per pair
// ---------------------------------------------------------------------------
__global__ void node_attn(const float* __restrict__ h, const float* __restrict__ a_s,
                          const float* __restrict__ a_d, float* __restrict__ al_s,
                          float* __restrict__ al_d, int n_nodes, int C) {
  const int wave = threadIdx.x >> 5, lane = threadIdx.x & 31;
  const int idx = blockIdx.x * 8 + wave;
  if (idx >= n_nodes * H) return;
  const int n = idx / H, hh = idx % H;
  const float* row = h + (size_t)n * H * C + hh * C;
  const float* as_ = a_s + hh * C;
  const float* ad_ = a_d + hh * C;
  float ss = 0.f, sd = 0.f;
  for (int c = lane; c < C; c += 32) {
    const float v = row[c];
    ss += v * as_[c];
    sd += v * ad_[c];
  }
  for (int o = 16; o > 0; o >>= 1) {
    ss += __shfl_xor(ss, o, 32);
    sd += __shfl_xor(sd, o, 32);
  }
  if (lane == 0) { al_s[idx] = ss; al_d[idx] = sd; }
}

// ---------------------------------------------------------------------------
// edge logits (leaky_relu 0.2) + segment max via flipped-uint atomicMax
// edge e < NE: real edge; e >= NE: self loop of node (e-NE) with ea=mean
// ---------------------------------------------------------------------------
__global__ void edge_logits(const int* __restrict__ ei, const float* __restrict__ ea,
                            const float* __restrict__ misc, int ceOff,
                            const float* __restrict__ al_s, const float* __restrict__ al_d,
                            float* __restrict__ logits, unsigned* __restrict__ mkey) {
  const int e = blockIdx.x * blockDim.x + threadIdx.x;
  if (e >= ET) return;
  int s, d; float eav;
  if (e < NE) { s = ei[e]; d = ei[NE + e]; eav = ea[e]; }
  else        { s = d = e - NE;            eav = misc[0]; }
#pragma unroll
  for (int hh = 0; hh < H; ++hh) {
    float v = al_s[s * H + hh] + al_d[d * H + hh] + eav * misc[ceOff + hh];
    v = v >= 0.f ? v : 0.2f * v;
    logits[(size_t)e * H + hh] = v;
    atomicMax(&mkey[d * H + hh], flipf(v));
  }
}

// p = exp(logit - m[dst]); denom[dst] += p  (p overwrites logits buffer)
__global__ void edge_softmax_p(const int* __restrict__ ei, float* __restrict__ logits,
                               const unsigned* __restrict__ mkey, float* __restrict__ denom) {
  const int e = blockIdx.x * blockDim.x + threadIdx.x;
  if (e >= ET) return;
  const int d = (e < NE) ? ei[NE + e] : (e - NE);
#pragma unroll
  for (int hh = 0; hh < H; ++hh) {
    float m = unflipf(mkey[d * H + hh]);
    if (!(fabsf(m) < 1e37f)) m = 0.f;       // isfinite guard (NaN/inf -> 0)
    const float p = expf(logits[(size_t)e * H + hh] - m);
    logits[(size_t)e * H + hh] = p;
    atomicAdd(&denom[d * H + hh], p);
  }
}

// out[dst] += h[src] * alpha   (alpha = p / (denom[dst] + 1e-16)); block per edge
__global__ void edge_aggregate(const int* __restrict__ ei, const float* __restrict__ p,
                               const float* __restrict__ denom, const float* __restrict__ hsrc,
                               float* __restrict__ out, int C) {
  const int e = blockIdx.x;
  int s, d;
  if (e < NE) { s = ei[e]; d = ei[NE + e]; }
  else        { s = d = e - NE; }
  const int HC = H * C;
  for (int t = threadIdx.x; t < HC; t += blockDim.x) {
    const int hh = t / C;
    const float alpha = p[(size_t)e * H + hh] / (denom[d * H + hh] + 1e-16f);
    atomicAdd(&out[(size_t)d * HC + t], hsrc[(size_t)s * HC + t] * alpha);
  }
}

// ---------------------------------------------------------------------------
// elementwise bias, batchnorm stats/apply, head-mean, pooling, MLP
// ---------------------------------------------------------------------------
__global__ void add_bias(float* __restrict__ x, const float* __restrict__ b,
                         size_t n, int F) {
  const size_t i = (size_t)blockIdx.x * blockDim.x + threadIdx.x;
  if (i >= n) return;
  x[i] += b[i % F];
}

__global__ void bn_stats(const float* __restrict__ x, float* __restrict__ stats,
                         int n, int F) {
  __shared__ float sh[256], sh2[256];
  const int f = blockIdx.x;
  float s = 0.f, s2 = 0.f;
  for (int i = threadIdx.x; i < n; i += blockDim.x) {
    const float v = x[(size_t)i * F + f];
    s += v; s2 += v * v;
  }
  sh[threadIdx.x] = s; sh2[threadIdx.x] = s2; __syncthreads();
  for (int o = 128; o > 0; o >>= 1) {
    if (threadIdx.x < o) { sh[threadIdx.x] += sh[threadIdx.x + o]; sh2[threadIdx.x] += sh2[threadIdx.x + o]; }
    __syncthreads();
  }
  if (threadIdx.x == 0) {
    const float mu = sh[0] / n;
    stats[f]     = mu;
    stats[F + f] = sh2[0] / n - mu * mu;   // biased variance (matches jnp var)
  }
}

__global__ void bn_elu(float* __restrict__ x, const float* __restrict__ stats,
                       const float* __restrict__ gamma, const float* __restrict__ beta,
                       size_t n, int F) {
  const size_t i = (size_t)blockIdx.x * blockDim.x + threadIdx.x;
  if (i >= n) return;
  const int f = (int)(i % F);
  const float v = gamma[f] * (x[i] - stats[f]) * rsqrtf(stats[F + f] + 1e-5f) + beta[f];
  x[i] = elu1(v);
}

__global__ void head_mean_bias(const float* __restrict__ g2, const float* __restrict__ b2,
                               float* __restrict__ z) {
  const int i = blockIdx.x * blockDim.x + threadIdx.x;   // node*C2
  if (i >= NN * C2) return;
  const int nn = i / C2, c = i % C2;
  float s = 0.f;
#pragma unroll
  for (int hh = 0; hh < H; ++hh) s += g2[(size_t)nn * HC2 + hh * C2 + c];
  z[i] = s * (1.f / H) + b2[c];
}

__global__ void pool(const float* __restrict__ z, const int* __restrict__ batch,
                     float* __restrict__ psum, float* __restrict__ pcnt) {
  const int nn = blockIdx.x;
  const int b = batch[nn];
  for (int c = threadIdx.x; c < C2; c += blockDim.x)
    atomicAdd(&psum[b * C2 + c], z[(size_t)nn * C2 + c]);
  if (threadIdx.x == 0) atomicAdd(&pcnt[b], 1.0f);
}

__global__ void mlp_head(const float* __restrict__ psum, const float* __restrict__ pcnt,
                         const float* __restrict__ fw1, const float* __restrict__ fb1,
                         const float* __restrict__ fw2, const float* __restrict__ fb2,
                         float* __restrict__ out) {
  __shared__ float hid[NG * 32];
  const int t = threadIdx.x;          // 1024 threads: (graph, hidden-unit)
  const int g = t >> 5, j = t & 31;
  const float inv = 1.f / fmaxf(pcnt[g], 1.0f);
  float s = fb1[j];
  for (int c = 0; c < C2; ++c) s += (psum[g * C2 + c] * inv) * fw1[c * 32 + j];
  hid[g * 32 + j] = elu1(s);
  __syncthreads();
  if (j == 0) {
    float o = fb2[0];
#pragma unroll
    for (int k = 0; k < 32; ++k) o += hid[g * 32 + k] * fw2[k];
    out[g] = o;
  }
}

// ---------------------------------------------------------------------------
// launch
// ---------------------------------------------------------------------------
extern "C" void kernel_launch(void* const* d_in, const int* in_sizes, int n_in,
                              void* d_out, int out_size, void* d_ws, size_t ws_size,
                              hipStream_t stream) {
  const float* x   = (const float*)d_in[0];
  const int*   ei  = (const int*)d_in[1];
  const int*   bat = (const int*)d_in[2];
  const float* ea  = (const float*)d_in[3];
  const float* W1  = (const float*)d_in[4];
  const float* as1 = (const float*)d_in[5];
  const float* ad1 = (const float*)d_in[6];
  const float* We1 = (const float*)d_in[7];
  const float* ae1 = (const float*)d_in[8];
  const float* b1  = (const float*)d_in[9];
  const float* g1  = (const float*)d_in[10];
  const float* be1 = (const float*)d_in[11];
  const float* W2  = (const float*)d_in[12];
  const float* as2 = (const float*)d_in[13];
  const float* ad2 = (const float*)d_in[14];
  const float* We2 = (const float*)d_in[15];
  const float* ae2 = (const float*)d_in[16];
  const float* b2  = (const float*)d_in[17];
  const float* g2  = (const float*)d_in[18];
  const float* be2 = (const float*)d_in[19];
  const float* fw1 = (const float*)d_in[20];
  const float* fb1 = (const float*)d_in[21];
  const float* fw2 = (const float*)d_in[22];
  const float* fb2 = (const float*)d_in[23];
  float* out = (float*)d_out;

  // workspace layout (floats)
  float* ws = (float*)d_ws;
  float*    hbuf   = ws;                                  // h1 [NN,512]; reused as h2 [NN,256]
  float*    gbuf   = hbuf  + (size_t)NN * HC1;            // layer1 agg -> BN/ELU -> x2 [NN,512]
  float*    g2out  = gbuf  + (size_t)NN * HC1;            // layer2 agg [NN,256]
  float*    z      = g2out + (size_t)NN * HC2;            // [NN,64]
  float*    logits = z     + (size_t)NN * C2;             // [ET,4] (later holds p)
  float*    al_s   = logits + (size_t)ET * H;             // [NN,4]
  float*    al_d   = al_s  + (size_t)NN * H;
  unsigned* mkey   = (unsigned*)(al_d + (size_t)NN * H);  // [NN,4]
  float*    denom  = (float*)mkey + (size_t)NN * H;       // [NN,4]
  float*    stats  = denom + (size_t)NN * H;              // [2*512]
  float*    psum   = stats + 1024;                        // [32,64]
  float*    pcnt   = psum  + NG * C2;                     // [32]
  float*    misc   = pcnt  + NG;                          // [0]=ea mean, [1..4]=ce1, [5..8]=ce2

  // ---- zero accumulators (replayed every call) ----
  hipMemsetAsync(misc,  0, 16 * sizeof(float), stream);
  hipMemsetAsync(gbuf,  0, (size_t)NN * HC1 * sizeof(float), stream);
  hipMemsetAsync(mkey,  0, (size_t)NN * H * sizeof(unsigned), stream);
  hipMemsetAsync(denom, 0, (size_t)NN * H * sizeof(float), stream);
  hipMemsetAsync(psum,  0, NG * C2 * sizeof(float), stream);
  hipMemsetAsync(pcnt,  0, NG * sizeof(float), stream);

  // ---- edge_attr mean + attention-edge coefficients ----
  reduce_sum<<<256, 256, 0, stream>>>(ea, misc, NE);
  finalize_mean<<<1, 1, 0, stream>>>(misc, NE);
  edge_coef<<<H, 128, 0, stream>>>(We1, ae1, misc + 1, C1);
  edge_coef<<<H, 128, 0, stream>>>(We2, ae2, misc + 5, C2);

  // ================= GAT layer 1 =================
  gemm_wmma_f32<<<dim3(NN / 16, HC1 / 256), 128, 0, stream>>>(x, W1, hbuf, NN, DIN, HC1);
  node_attn<<<(NN * H + 7) / 8, 256, 0, stream>>>(hbuf, as1, ad1, al_s, al_d, NN, C1);
  edge_logits<<<(ET + 255) / 256, 256, 0, stream>>>(ei, ea, misc, 1, al_s, al_d, logits, mkey);
  edge_softmax_p<<<(ET + 255) / 256, 256, 0, stream>>>(ei, logits, mkey, denom);
  edge_aggregate<<<ET, 256, 0, stream>>>(ei, logits, denom, hbuf, gbuf, C1);
  add_bias<<<(int)(((size_t)NN * HC1 + 255) / 256), 256, 0, stream>>>(gbuf, b1, (size_t)NN * HC1, HC1);
  bn_stats<<<HC1, 256, 0, stream>>>(gbuf, stats, NN, HC1);
  bn_elu<<<(int)(((size_t)NN * HC1 + 255) / 256), 256, 0, stream>>>(gbuf, stats, g1, be1, (size_t)NN * HC1, HC1);

  // ================= GAT layer 2 =================
  hipMemsetAsync(mkey,  0, (size_t)NN * H * sizeof(unsigned), stream);
  hipMemsetAsync(denom, 0, (size_t)NN * H * sizeof(float), stream);
  hipMemsetAsync(g2out, 0, (size_t)NN * HC2 * sizeof(float), stream);

  gemm_wmma_f32<<<dim3(NN / 16, 1), 128, 0, stream>>>(gbuf, W2, hbuf, NN, HC1, HC2);
  node_attn<<<(NN * H + 7) / 8, 256, 0, stream>>>(hbuf, as2, ad2, al_s, al_d, NN, C2);
  edge_logits<<<(ET + 255) / 256, 256, 0, stream>>>(ei, ea, misc, 5, al_s, al_d, logits, mkey);
  edge_softmax_p<<<(ET + 255) / 256, 256, 0, stream>>>(ei, logits, mkey, denom);
  edge_aggregate<<<ET, 256, 0, stream>>>(ei, logits, denom, hbuf, g2out, C2);
  head_mean_bias<<<(NN * C2 + 255) / 256, 256, 0, stream>>>(g2out, b2, z);
  bn_stats<<<C2, 256, 0, stream>>>(z, stats, NN, C2);
  bn_elu<<<(int)(((size_t)NN * C2 + 255) / 256), 256, 0, stream>>>(z, stats, g2, be2, (size_t)NN * C2, C2);

  // ================= pool + MLP =================
  pool<<<NN, 64, 0, stream>>>(z, bat, psum, pcnt);
  mlp_head<<<1, 1024, 0, stream>>>(psum, pcnt, fw1, fb1, fw2, fb2, out);
}